// Group_41472204210680
// MI455X (gfx1250) — compile-verified
//
#include <hip/hip_runtime.h>

// Problem constants (from reference)
#define B_  32      // batches
#define N_  8192    // points per batch
#define G_  512     // NUM_GROUP (FPS centers)
#define K_  32      // GROUP_SIZE (kNN)
#define CH_ 256     // kNN point chunk size

typedef float v2f __attribute__((ext_vector_type(2)));
typedef float v8f __attribute__((ext_vector_type(8)));

// ---------------------------------------------------------------------------
// Kernel A: farthest point sampling. One workgroup per batch, 1024 threads.
// Points + running min-distance live entirely in VGPRs (8 per thread).
// The winner's coordinates ride along the argmax reductions, so the 512-step
// serial loop has ZERO global loads: pure VALU + shuffles + 2 barriers.
// ---------------------------------------------------------------------------
__global__ __launch_bounds__(1024) void fps_kernel(const float* __restrict__ xyz,
                                                   float* __restrict__ center) {
  const int b = blockIdx.x;
  const int t = threadIdx.x;
  const float* xb = xyz + (size_t)b * 3 * N_;
  const float* yb = xb + N_;
  const float* zb = xb + 2 * N_;

  float px[8], py[8], pz[8], md[8];
#pragma unroll
  for (int i = 0; i < 8; ++i) {
    int n = i * 1024 + t;            // strided ownership -> coalesced loads
    px[i] = xb[n]; py[i] = yb[n]; pz[i] = zb[n];
    md[i] = 1e10f;
  }

  __shared__ float wval[32], wx[32], wy[32], wz[32];
  __shared__ int   widx[32];
  __shared__ float s_cx, s_cy, s_cz;

  float* cb = center + (size_t)b * G_ * 3;

  // initial centroid = point 0 (uniform broadcast load, once)
  float cx = xb[0], cy = yb[0], cz = zb[0];

  const int lane = t & 31, w = t >> 5;

  for (int it = 0; it < G_; ++it) {
    if (t == 0) { cb[it * 3 + 0] = cx; cb[it * 3 + 1] = cy; cb[it * 3 + 2] = cz; }

    float bv = -1.0f, bx = 0.f, by = 0.f, bz = 0.f; int bi = 0;
#pragma unroll
    for (int i = 0; i < 8; ++i) {
      float dx = px[i] - cx, dy = py[i] - cy, dz = pz[i] - cz;
      float d = dx * dx + dy * dy + dz * dz;
      md[i] = fminf(md[i], d);
      int n = i * 1024 + t;
      if (md[i] > bv || (md[i] == bv && n < bi)) {
        bv = md[i]; bi = n; bx = px[i]; by = py[i]; bz = pz[i];
      }
    }
    // wave32 butterfly argmax (max value, min index on tie), coords ride along
#pragma unroll
    for (int off = 16; off >= 1; off >>= 1) {
      float ov = __shfl_xor(bv, off, 32);
      int   oi = __shfl_xor(bi, off, 32);
      float ox = __shfl_xor(bx, off, 32);
      float oy = __shfl_xor(by, off, 32);
      float oz = __shfl_xor(bz, off, 32);
      if (ov > bv || (ov == bv && oi < bi)) { bv = ov; bi = oi; bx = ox; by = oy; bz = oz; }
    }
    if (lane == 0) { wval[w] = bv; widx[w] = bi; wx[w] = bx; wy[w] = by; wz[w] = bz; }
    __syncthreads();
    if (t < 32) {                    // wave 0 reduces the 32 wave winners
      bv = wval[t]; bi = widx[t]; bx = wx[t]; by = wy[t]; bz = wz[t];
#pragma unroll
      for (int off = 16; off >= 1; off >>= 1) {
        float ov = __shfl_xor(bv, off, 32);
        int   oi = __shfl_xor(bi, off, 32);
        float ox = __shfl_xor(bx, off, 32);
        float oy = __shfl_xor(by, off, 32);
        float oz = __shfl_xor(bz, off, 32);
        if (ov > bv || (ov == bv && oi < bi)) { bv = ov; bi = oi; bx = ox; by = oy; bz = oz; }
      }
      if (t == 0) { s_cx = bx; s_cy = by; s_cz = bz; }
    }
    __syncthreads();
    cx = s_cx; cy = s_cy; cz = s_cz;
  }
}

// ---------------------------------------------------------------------------
// Kernel B: kNN + gather + center-subtract.
// Grid: (G/16, B); block 256 (8 waves). Each WG owns 16 centers.
// Points stream through LDS in 256-point chunks; each wave computes two
// 16x16 distance tiles per chunk via V_WMMA_F32_16X16X4_F32
// (A = centers 16x4 [x,y,z,0], B = points 4x16). Next chunk is prefetched
// (global_prefetch_b8). Running sorted top-32 per center in LDS; chunks
// that cannot beat the current 32nd-best are skipped via ballot.
// ---------------------------------------------------------------------------
__global__ __launch_bounds__(256) void knn_kernel(const float* __restrict__ xyz,
                                                  const float* __restrict__ center,
                                                  float* __restrict__ nbh) {
  const int gt = blockIdx.x;   // center tile within batch (0..31)
  const int b  = blockIdx.y;   // batch
  const int t  = threadIdx.x;
  const int lane = t & 31;
  const int w = t >> 5;        // wave 0..7

  const float* xb = xyz + (size_t)b * 3 * N_;
  const float* yb = xb + N_;
  const float* zb = xb + 2 * N_;
  const float* cb = center + (size_t)b * G_ * 3 + (size_t)gt * 16 * 3;

  __shared__ float sx[CH_], sy[CH_], sz[CH_], spn[CH_];
  __shared__ float cdx[16], cdy[16], cdz[16], cnrm[16];
  __shared__ float dtile[16][CH_ + 4];   // padded stride vs. bank conflicts
  __shared__ float topd[16][K_];
  __shared__ int   topi[16][K_];
  __shared__ float newd[8][K_];
  __shared__ int   newi[8][K_];

  if (t < 16) {
    float x = cb[t * 3 + 0], y = cb[t * 3 + 1], z = cb[t * 3 + 2];
    cdx[t] = x; cdy[t] = y; cdz[t] = z;
    cnrm[t] = x * x + y * y + z * z;
  }
  for (int i = t; i < 16 * K_; i += 256) {
    (&topd[0][0])[i] = 1e30f;
    (&topi[0][0])[i] = 0;
  }
  __syncthreads();

  // A operand: 16x4 f32 centers tile.  ISA layout: lanes 0-15 hold K=0,1 in
  // the two VGPRs; lanes 16-31 hold K=2,3.  K=3 column padded with 0.
  v2f A;
  {
    int m = lane & 15;
    if (lane < 16) { A.x = cdx[m]; A.y = cdy[m]; }
    else           { A.x = cdz[m]; A.y = 0.0f;   }
  }

  const int NCHUNK = N_ / CH_;
  for (int ch = 0; ch < NCHUNK; ++ch) {
    const int base = ch * CH_;

    // prefetch next chunk's cachelines (x,y,z) -> global_prefetch_b8
    if (ch + 1 < NCHUNK && t < 24) {
      const float* pf = (t < 8) ? xb : (t < 16) ? yb : zb;
      __builtin_prefetch(pf + base + CH_ + (t & 7) * 32, 0, 0);
    }

    // stage 256 points + their squared norms
    {
      float x = xb[base + t], y = yb[base + t], z = zb[base + t];
      sx[t] = x; sy[t] = y; sz[t] = z;
      spn[t] = x * x + y * y + z * z;
    }
    __syncthreads();

    // two 16-point sub-tiles per wave: cols w*16 and 128 + w*16
#pragma unroll
    for (int half = 0; half < 2; ++half) {
      const int col0 = half * 128 + w * 16;
      // B operand: 4x16 f32. Rows striped across lanes: lanes 0-15 -> rows
      // K=0,1; lanes 16-31 -> rows K=2,3 (pad row 3 = 0).
      v2f Bm;
      {
        int n = col0 + (lane & 15);
        if (lane < 16) { Bm.x = sx[n]; Bm.y = sy[n]; }
        else           { Bm.x = sz[n]; Bm.y = 0.0f;  }
      }
      v8f C = {};
      v8f D = __builtin_amdgcn_wmma_f32_16x16x4_f32(
          /*neg_a=*/false, A, /*neg_b=*/false, Bm,
          /*c_mod=*/(short)0, C, /*reuse_a=*/false, /*reuse_b=*/false);

      // d^2 = |c|^2 + |p|^2 - 2 c.p  -> LDS distance tile
      int n = lane & 15;
      int col = col0 + n;
      int mbase = (lane < 16) ? 0 : 8;
      float pn = spn[col];
#pragma unroll
      for (int r = 0; r < 8; ++r) {
        int m = mbase + r;
        dtile[m][col] = cnrm[m] + pn - 2.0f * D[r];
      }
    }
    __syncthreads();

    // each wave merges this chunk into the top-32 of its two centers
#pragma unroll
    for (int s = 0; s < 2; ++s) {
      int c = w * 2 + s;
      volatile float* tprow = topd[c];
      volatile int*   tirow = topi[c];
      volatile float* drow  = dtile[c];

      float thr = topd[c][K_ - 1];        // current 32nd-best (list sorted)
      bool anyc = false;
#pragma unroll
      for (int q = 0; q < 8; ++q)
        if (dtile[c][lane * 8 + q] < thr) anyc = true;

      if (__any(anyc)) {
        // 32 argmin passes over 32 running + 256 chunk candidates.
        for (int j = 0; j < K_; ++j) {
          float v = tprow[lane]; int id = lane;          // ids 0..31 = running
#pragma unroll
          for (int q = 0; q < 8; ++q) {                  // ids 32..287 = chunk
            int cc = lane * 8 + q;
            float dv = drow[cc];
            if (dv < v || (dv == v && (32 + cc) < id)) { v = dv; id = 32 + cc; }
          }
#pragma unroll
          for (int off = 16; off >= 1; off >>= 1) {
            float ov = __shfl_xor(v, off, 32);
            int  oid = __shfl_xor(id, off, 32);
            if (ov < v || (ov == v && oid < id)) { v = ov; id = oid; }
          }
          if (lane == 0) {
            newd[w][j] = v;
            newi[w][j] = (id < 32) ? tirow[id] : (base + (id - 32));
            if (id < 32) tprow[id] = 1e30f;              // invalidate source
            else         drow[id - 32] = 1e30f;
          }
        }
        tprow[lane] = newd[w][lane];                     // sorted ascending
        tirow[lane] = newi[w][lane];
      }
    }
    __syncthreads();
  }

  // gather neighborhoods and subtract center
#pragma unroll
  for (int s = 0; s < 2; ++s) {
    int c = w * 2 + s;
    int gidx = topi[c][lane];
    float ox = xb[gidx] - cdx[c];
    float oy = yb[gidx] - cdy[c];
    float oz = zb[gidx] - cdz[c];
    size_t o = (((size_t)b * G_ + (size_t)(gt * 16 + c)) * K_ + lane) * 3;
    nbh[o + 0] = ox; nbh[o + 1] = oy; nbh[o + 2] = oz;
  }
}

// ---------------------------------------------------------------------------
extern "C" void kernel_launch(void* const* d_in, const int* in_sizes, int n_in,
                              void* d_out, int out_size, void* d_ws, size_t ws_size,
                              hipStream_t stream) {
  (void)in_sizes; (void)n_in; (void)out_size; (void)d_ws; (void)ws_size;
  const float* xyz = (const float*)d_in[0];       // [32, 3, 8192] f32
  float* out = (float*)d_out;
  float* nbh = out;                                // [32,512,32,3]
  float* ctr = out + (size_t)B_ * G_ * K_ * 3;     // [32,512,3]

  fps_kernel<<<dim3(B_), dim3(1024), 0, stream>>>(xyz, ctr);
  knn_kernel<<<dim3(G_ / 16, B_), dim3(256), 0, stream>>>(xyz, ctr, nbh);
}